// GCN_4140348474048
// MI455X (gfx1250) — compile-verified
//
#include <hip/hip_runtime.h>
#include <hip/hip_bf16.h>

typedef __attribute__((ext_vector_type(2))) float v2f;
typedef __attribute__((ext_vector_type(8))) float v8f;

#define NNODES 50000
#define CH 128
#define NTILES (NNODES / 16)   // 3125, exact
#define BN_EPS 1e-5f

// ---------------------------------------------------------------------------
// Zero-fill (float4 stores)
// ---------------------------------------------------------------------------
__global__ void __launch_bounds__(256)
k_zero(float4* __restrict__ p, int n4) {
    int i = blockIdx.x * blockDim.x + threadIdx.x;
    int stride = gridDim.x * blockDim.x;
    float4 z = make_float4(0.f, 0.f, 0.f, 0.f);
    for (; i < n4; i += stride) p[i] = z;
}

// ---------------------------------------------------------------------------
// Fused  h = x @ W^T + b ;  BatchNorm1d(N) over (batch, channel) ;  exact GELU
// One wave handles a 16-node tile for BOTH batch elements (BN couples them):
//   16 v8f accumulators = [2 batches][8 column tiles of 16 channels].
// A-frag (16x4 f32):  lane<16 -> {x[m=lane][k],x[m=lane][k+1]},
//                     lane>=16 -> {x[m=lane-16][k+2], x[m=lane-16][k+3]}
// B-frag (4x16 f32):  Btile[k][n] = W[n][k], same half-split on K.
// C/D 16x16 f32: element v of lane L -> row M = v + 8*(L>=16), col N = L&15.
// ---------------------------------------------------------------------------
__global__ void __launch_bounds__(256)
k_gemm_bn_gelu(const float* __restrict__ x,    // [2,N,128]
               const float* __restrict__ W,    // [128,128] row-major (out,f)
               const float* __restrict__ bias, // [128]
               const float* __restrict__ g,    // [N]
               const float* __restrict__ be,   // [N]
               float* __restrict__ out)        // [2,N,128]
{
    __shared__ float sW[CH * CH];              // 64 KB of W in LDS
    {
        const float4* src = (const float4*)W;
        float4* dst = (float4*)sW;
        for (int i = threadIdx.x; i < (CH * CH) / 4; i += 256) dst[i] = src[i];
    }
    __syncthreads();

    const int wave = threadIdx.x >> 5;
    const int lane = threadIdx.x & 31;
    const int tile = blockIdx.x * 8 + wave;
    if (tile >= NTILES) return;

    const int node_base = tile * 16;
    const int half = lane >> 4;    // 0: lanes 0-15, 1: lanes 16-31
    const int idx  = lane & 15;

    v8f acc[2][8];
    {
        v8f z = {};
        for (int b = 0; b < 2; ++b)
            for (int t = 0; t < 8; ++t) acc[b][t] = z;
    }

    const float* x0 = x + (size_t)node_base * CH;
    const float* x1 = x + (size_t)NNODES * CH + (size_t)node_base * CH;
    const int kk0 = half * 2;

    for (int k = 0; k < CH; k += 4) {
        const int kk = k + kk0;
        v2f a0 = *(const v2f*)(x0 + idx * CH + kk);
        v2f a1 = *(const v2f*)(x1 + idx * CH + kk);
#pragma unroll
        for (int t = 0; t < 8; ++t) {
            v2f bb = *(const v2f*)(&sW[(t * 16 + idx) * CH + kk]);
            acc[0][t] = __builtin_amdgcn_wmma_f32_16x16x4_f32(
                false, a0, false, bb, (short)0, acc[0][t], false, false);
            acc[1][t] = __builtin_amdgcn_wmma_f32_16x16x4_f32(
                false, a1, false, bb, (short)0, acc[1][t], false, false);
        }
    }

    // bias (per output channel n = t*16 + idx, same for all rows v)
#pragma unroll
    for (int t = 0; t < 8; ++t) {
        float bv = bias[t * 16 + idx];
#pragma unroll
        for (int v = 0; v < 8; ++v) {
            acc[0][t][v] += bv;
            acc[1][t][v] += bv;
        }
    }

    // BatchNorm over (batch, channel): 256 values per node row, then GELU.
    const float inv256 = 1.0f / 256.0f;
#pragma unroll
    for (int v = 0; v < 8; ++v) {
        float s = 0.f, sq = 0.f;
#pragma unroll
        for (int b = 0; b < 2; ++b)
#pragma unroll
            for (int t = 0; t < 8; ++t) {
                float h = acc[b][t][v];
                s += h; sq += h * h;
            }
        // reduce across the 16 lanes of this half (one matrix row)
#pragma unroll
        for (int m = 1; m <= 8; m <<= 1) {
            s  += __shfl_xor(s,  m, 32);
            sq += __shfl_xor(sq, m, 32);
        }
        const int node = node_base + v + 8 * half;
        float mu  = s * inv256;
        float var = sq * inv256 - mu * mu;
        float rs  = rsqrtf(var + BN_EPS);
        float gn  = g[node];
        float scale = gn * rs;
        float shift = be[node] - mu * scale;
#pragma unroll
        for (int b = 0; b < 2; ++b)
#pragma unroll
            for (int t = 0; t < 8; ++t) {
                float y = acc[b][t][v] * scale + shift;
                acc[b][t][v] = 0.5f * y * (1.0f + erff(y * 0.70710678118654752f));
            }
    }

    // store activations
#pragma unroll
    for (int b = 0; b < 2; ++b) {
        float* o = out + (size_t)b * NNODES * CH + (size_t)node_base * CH;
#pragma unroll
        for (int v = 0; v < 8; ++v) {
            const int rofs = (v + 8 * half) * CH;
#pragma unroll
            for (int t = 0; t < 8; ++t)
                o[rofs + t * 16 + idx] = acc[b][t][v];
        }
    }
}

// ---------------------------------------------------------------------------
// Message passing: out[b, row[e], :] += a[b, col[e], :] * norm[e]
// One wave per edge; each lane: 4 channels (float4 gather + 4 f32 atomics).
// ---------------------------------------------------------------------------
__global__ void __launch_bounds__(256)
k_scatter(const float* __restrict__ a,
          const int* __restrict__ row,
          const int* __restrict__ col,
          const float* __restrict__ norm,
          float* __restrict__ out, int E)
{
    int e = (blockIdx.x * blockDim.x + threadIdx.x) >> 5;
    int lane = threadIdx.x & 31;
    if (e >= E) return;
    int r = row[e];
    int c = col[e];
    float nm = norm[e];
    int c0 = lane * 4;
#pragma unroll
    for (int b = 0; b < 2; ++b) {
        const float4 src = *(const float4*)(a + (size_t)b * NNODES * CH + (size_t)c * CH + c0);
        float* dst = out + (size_t)b * NNODES * CH + (size_t)r * CH + c0;
        __hip_atomic_fetch_add(dst + 0, src.x * nm, __ATOMIC_RELAXED, __HIP_MEMORY_SCOPE_AGENT);
        __hip_atomic_fetch_add(dst + 1, src.y * nm, __ATOMIC_RELAXED, __HIP_MEMORY_SCOPE_AGENT);
        __hip_atomic_fetch_add(dst + 2, src.z * nm, __ATOMIC_RELAXED, __HIP_MEMORY_SCOPE_AGENT);
        __hip_atomic_fetch_add(dst + 3, src.w * nm, __ATOMIC_RELAXED, __HIP_MEMORY_SCOPE_AGENT);
    }
}

// ---------------------------------------------------------------------------
extern "C" void kernel_launch(void* const* d_in, const int* in_sizes, int n_in,
                              void* d_out, int out_size, void* d_ws, size_t ws_size,
                              hipStream_t stream)
{
    const float* x   = (const float*)d_in[0];
    const int*   ei  = (const int*)d_in[1];   // [2,E]
    const float* nrm = (const float*)d_in[2]; // [E]
    const float* W1  = (const float*)d_in[3];
    const float* b1  = (const float*)d_in[4];
    const float* g1  = (const float*)d_in[5];
    const float* be1 = (const float*)d_in[6];
    const float* W2  = (const float*)d_in[7];
    const float* b2  = (const float*)d_in[8];
    const float* g2  = (const float*)d_in[9];
    const float* be2 = (const float*)d_in[10];

    const int E = in_sizes[2];          // norm has E elements
    const int* row = ei;                // edge_index[0]
    const int* col = ei + E;            // edge_index[1]

    const size_t elems = (size_t)2 * NNODES * CH;   // 12.8M floats
    float* bufA = (float*)d_ws;                      // activations
    float* bufH = bufA + elems;                      // layer-1 output
    float* outF = (float*)d_out;

    dim3 blk(256);
    dim3 gGemm((NTILES + 7) / 8);
    dim3 gScat((E + 7) / 8);
    dim3 gZero(1024);
    const int n4 = (int)(elems / 4);

    // ---- layer 1 ----
    k_gemm_bn_gelu<<<gGemm, blk, 0, stream>>>(x, W1, b1, g1, be1, bufA);
    k_zero<<<gZero, blk, 0, stream>>>((float4*)bufH, n4);
    k_scatter<<<gScat, blk, 0, stream>>>(bufA, row, col, nrm, bufH, E);

    // ---- layer 2 ----
    k_gemm_bn_gelu<<<gGemm, blk, 0, stream>>>(bufH, W2, b2, g2, be2, bufA);
    k_zero<<<gZero, blk, 0, stream>>>((float4*)outF, n4);
    k_scatter<<<gScat, blk, 0, stream>>>(bufA, row, col, nrm, outF, E);
}